// Fusion_Block_39264591020441
// MI455X (gfx1250) — compile-verified
//
#include <hip/hip_runtime.h>
#include <math.h>

// Shapes
#define B_    64
#define L_    128
#define D_    64
#define K_    64
#define O_    90
#define PL_   16
#define PRED_ 720
#define DM_   512
#define OPL_  1440   // O*PL
#define EPS_  1e-5f

typedef float v2f __attribute__((ext_vector_type(2)));
typedef float v8f __attribute__((ext_vector_type(8)));
typedef int   v4i __attribute__((vector_size(16)));

// D = A(16x4 f32) * B(4x16 f32) + C(16x16 f32)
static __device__ __forceinline__ v8f wmma4(v2f a, v2f b, v8f c) {
  return __builtin_amdgcn_wmma_f32_16x16x4_f32(false, a, false, b, (short)0, c,
                                               false, false);
}

// ---- global -> LDS 16B move: async (ASYNCcnt) when available, else via VGPR
#if defined(__gfx1250__) &&                                                    \
    __has_builtin(__builtin_amdgcn_global_load_async_to_lds_b128) &&           \
    __has_builtin(__builtin_amdgcn_s_wait_asynccnt)
#define ASYNC_LDS 1
#else
#define ASYNC_LDS 0
#endif

static __device__ __forceinline__ void g2lds16(const float* __restrict__ g,
                                               float* l) {
#if ASYNC_LDS
  __builtin_amdgcn_global_load_async_to_lds_b128(
      (__attribute__((address_space(1))) v4i*)g,
      (__attribute__((address_space(3))) v4i*)l, 0, 0);
#else
  const float4 v = *(const float4*)g;
  l[0] = v.x; l[1] = v.y; l[2] = v.z; l[3] = v.w;
#endif
}

static __device__ __forceinline__ void lds_async_wait() {
#if ASYNC_LDS
  __builtin_amdgcn_s_wait_asynccnt(0);
#endif
}

// ---------------- Kernel 1: per-(b,d) cross attention -> V (O*PL per head) ---
// LDS layout (floats); row pads keep 16B alignment + conflict-free frag reads.
#define K1_XD 0                        // [128][68]
#define K1_YD (K1_XD + 128 * 68)       // [96][68]
#define K1_S  (K1_YD + 96 * 68)        // [128][96]
#define K1_XP (K1_S + 128 * 96)        // [128][16]
#define K1_TOT (K1_XP + 128 * 16)      // 29568 floats = 118272 B

__global__ __launch_bounds__(256) void attn_kernel(
    const float* __restrict__ x, const float* __restrict__ x_date,
    const float* __restrict__ y_date, float* __restrict__ vout) {
  extern __shared__ float sm[];
  const int bd   = blockIdx.x;
  const int b    = bd >> 6;
  const int d    = bd & 63;
  const int tid  = threadIdx.x;
  const int lane = tid & 31;
  const int wave = tid >> 5;
  const int half = lane >> 4;
  const int l16  = lane & 15;

  // Stage x_date slice (L x K), row stride D*K (async -> LDS)
  const float* xd = x_date + (size_t)b * (L_ * D_ * K_) + (size_t)d * K_;
  for (int t = tid; t < 128 * 16; t += 256) {
    const int l = t >> 4, j = t & 15;
    g2lds16(xd + (size_t)l * (D_ * K_) + j * 4, sm + K1_XD + l * 68 + j * 4);
  }
  // Stage y_date slice (O x K)
  const float* yd = y_date + (size_t)b * (O_ * D_ * K_) + (size_t)d * K_;
  for (int t = tid; t < O_ * 16; t += 256) {
    const int o = t >> 4, j = t & 15;
    g2lds16(yd + (size_t)o * (D_ * K_) + j * 4, sm + K1_YD + o * 68 + j * 4);
  }
  // Zero-pad rows 90..95
  for (int t = O_ * 16 + tid; t < 96 * 16; t += 256) {
    const int o = t >> 4, j = t & 15;
    float* r = sm + K1_YD + o * 68 + j * 4;
    r[0] = 0.f; r[1] = 0.f; r[2] = 0.f; r[3] = 0.f;
  }
  // Stage x slice (L x PL), contiguous
  const float* xp = x + (size_t)bd * (L_ * PL_);
  for (int t = tid; t < 512; t += 256) g2lds16(xp + t * 4, sm + K1_XP + t * 4);
  lds_async_wait();
  __syncthreads();

  // Scores: S[l][o] = scale * sum_k Xd[l][k] * Yd[o][k]
  {
    const int mt = wave;  // 8 waves = 8 L-tiles
    const float* arow = sm + K1_XD + (mt * 16 + l16) * 68;
    for (int nt = 0; nt < 6; ++nt) {
      const float* brow = sm + K1_YD + (nt * 16 + l16) * 68;
      v8f acc = {};
      #pragma unroll
      for (int kk = 0; kk < 64; kk += 4) {
        const int c0 = kk + (half ? 2 : 0);
        v2f a, bb;
        a.x = arow[c0]; a.y = arow[c0 + 1];
        bb.x = brow[c0]; bb.y = brow[c0 + 1];
        acc = wmma4(a, bb, acc);
      }
      #pragma unroll
      for (int r = 0; r < 8; ++r) {
        const int m = mt * 16 + r + 8 * half;
        sm[K1_S + m * 96 + nt * 16 + l16] = acc[r] * 0.125f;  // 1/sqrt(64)
      }
    }
  }
  __syncthreads();

  // Softmax over l for each column o < 90
  if (tid < O_) {
    const int o = tid;
    float mx = -3.4e38f;
    for (int l = 0; l < 128; ++l) mx = fmaxf(mx, sm[K1_S + l * 96 + o]);
    float sum = 0.f;
    for (int l = 0; l < 128; ++l) {
      const float e = __expf(sm[K1_S + l * 96 + o] - mx);
      sm[K1_S + l * 96 + o] = e;
      sum += e;
    }
    const float inv = 1.f / sum;
    for (int l = 0; l < 128; ++l) sm[K1_S + l * 96 + o] *= inv;
  }
  __syncthreads();

  // V[o][p] = sum_l A[l][o] * Xp[l][p]   (A^T @ Xp), O padded to 96
  if (wave < 6) {
    const int mt = wave;
    v8f acc = {};
    #pragma unroll 8
    for (int kk = 0; kk < 128; kk += 4) {
      const int c0 = kk + (half ? 2 : 0);
      v2f a, bb;
      a.x = sm[K1_S + c0 * 96 + mt * 16 + l16];
      a.y = sm[K1_S + (c0 + 1) * 96 + mt * 16 + l16];
      bb.x = sm[K1_XP + c0 * 16 + l16];
      bb.y = sm[K1_XP + (c0 + 1) * 16 + l16];
      acc = wmma4(a, bb, acc);
    }
    float* vb = vout + (size_t)bd * OPL_;
    #pragma unroll
    for (int r = 0; r < 8; ++r) {
      const int o = mt * 16 + r + 8 * half;
      if (o < O_) vb[o * 16 + l16] = acc[r];
    }
  }
}

// ---------------- Kernel 2: h = V @ w_lin^T + b_lin, BN1 (double-buffered) --
static __device__ __forceinline__ void k2_load_tile(
    const float* __restrict__ vflat, const float* __restrict__ w_lin,
    float* At, float* Bt, int gm0, int gn0, int kt, int tid) {
  for (int t = tid; t < 768; t += 384) {
    const int m = t / 12, j = t % 12;
    g2lds16(vflat + (size_t)(gm0 + m) * OPL_ + kt + j * 4, At + m * 52 + j * 4);
  }
  for (int t = tid; t < 576; t += 384) {
    const int n = t / 12, j = t % 12;
    g2lds16(w_lin + (size_t)(gn0 + n) * OPL_ + kt + j * 4, Bt + n * 52 + j * 4);
  }
}

__global__ __launch_bounds__(384) void lin_bn1_kernel(
    const float* __restrict__ vflat, const float* __restrict__ w_lin,
    const float* __restrict__ b_lin, const float* __restrict__ g1,
    const float* __restrict__ be1, const float* __restrict__ m1,
    const float* __restrict__ v1, float* __restrict__ hout) {
  __shared__ float At[2][64 * 52];
  __shared__ float Bt[2][48 * 52];
  const int tid  = threadIdx.x;
  const int lane = tid & 31;
  const int wave = tid >> 5;
  const int half = lane >> 4;
  const int l16  = lane & 15;
  const int gm0  = blockIdx.x * 64;
  const int gn0  = blockIdx.y * 48;
  const int mt   = wave & 3;   // 4 M-tiles
  const int nt   = wave >> 2;  // 3 N-tiles

  k2_load_tile(vflat, w_lin, At[0], Bt[0], gm0, gn0, 0, tid);
  v8f acc = {};
  for (int it = 0; it < 30; ++it) {
    const int buf = it & 1;
    lds_async_wait();
    __syncthreads();  // tile `it` resident; all waves done reading buf^1
    if (it + 1 < 30)
      k2_load_tile(vflat, w_lin, At[buf ^ 1], Bt[buf ^ 1], gm0, gn0,
                   (it + 1) * 48, tid);
    const float* arow = &At[buf][(mt * 16 + l16) * 52];
    const float* brow = &Bt[buf][(nt * 16 + l16) * 52];
    #pragma unroll
    for (int kk = 0; kk < 48; kk += 4) {
      const int c0 = kk + (half ? 2 : 0);
      v2f a, bb;
      a.x = arow[c0]; a.y = arow[c0 + 1];
      bb.x = brow[c0]; bb.y = brow[c0 + 1];
      acc = wmma4(a, bb, acc);
    }
  }
  const int n = gn0 + nt * 16 + l16;
  const float bl = b_lin[n];
  #pragma unroll
  for (int r = 0; r < 8; ++r) {
    const int g = gm0 + mt * 16 + r + 8 * half;
    const int d = g & 63;
    const float s = g1[d] * rsqrtf(v1[d] + EPS_);
    hout[(size_t)g * PRED_ + n] = (acc[r] + bl - m1[d]) * s + be1[d];
  }
}

// ---------------- Kernel 3: FFN (conv1x1) + residual + BN2 + transpose ------
#define K3_WS 0                          // max(512*68, 64*516) = 34816
#define K3_T1 34816                      // [512][48]
#define K3_HT (34816 + 512 * 48)         // [64][52]
#define K3_TOT (K3_HT + 64 * 52)         // 62720 floats = 250880 B

__global__ __launch_bounds__(384) void ffn_bn2_kernel(
    const float* __restrict__ h, const float* __restrict__ w_c1,
    const float* __restrict__ b_c1, const float* __restrict__ w_c2,
    const float* __restrict__ b_c2, const float* __restrict__ g2,
    const float* __restrict__ be2, const float* __restrict__ m2,
    const float* __restrict__ v2, float* __restrict__ out) {
  extern __shared__ float sm[];
  const int tid  = threadIdx.x;
  const int lane = tid & 31;
  const int wave = tid >> 5;
  const int half = lane >> 4;
  const int l16  = lane & 15;
  const int b    = blockIdx.x;
  const int p0   = blockIdx.y * 48;

  // Stage W1 (512x64), pad-68 rows; H tile (64x48), pad-52 rows (async)
  for (int t = tid; t < 512 * 16; t += 384) {
    const int e = t >> 4, j = t & 15;
    g2lds16(w_c1 + e * 64 + j * 4, sm + K3_WS + e * 68 + j * 4);
  }
  for (int t = tid; t < 64 * 12; t += 384) {
    const int dd = t / 12, j = t % 12;
    g2lds16(h + ((size_t)b * 64 + dd) * PRED_ + p0 + j * 4,
            sm + K3_HT + dd * 52 + j * 4);
  }
  // Prefetch W2 (used after GEMM1) into cache (global_prefetch_b8)
  for (int t = tid; t < 512; t += 384) __builtin_prefetch(w_c2 + t * 64, 0, 1);
  lds_async_wait();
  __syncthreads();

  // GEMM1: T1[e][p] = gelu( sum_d W1[e][d]*Ht[d][p] + b_c1[e] )
  // 32 M-tiles x 3 N-tiles = 96 tiles, 12 waves * 8
  for (int i = 0; i < 8; ++i) {
    const int t  = wave * 8 + i;
    const int mt = t / 3, nt = t % 3;
    const float* arow = sm + K3_WS + (mt * 16 + l16) * 68;
    v8f acc = {};
    #pragma unroll
    for (int kk = 0; kk < 64; kk += 4) {
      const int c0 = kk + (half ? 2 : 0);
      v2f a, bb;
      a.x = arow[c0]; a.y = arow[c0 + 1];
      bb.x = sm[K3_HT + c0 * 52 + nt * 16 + l16];
      bb.y = sm[K3_HT + (c0 + 1) * 52 + nt * 16 + l16];
      acc = wmma4(a, bb, acc);
    }
    #pragma unroll
    for (int r = 0; r < 8; ++r) {
      const int e = mt * 16 + r + 8 * half;
      const float xv = acc[r] + b_c1[e];
      sm[K3_T1 + e * 48 + nt * 16 + l16] =
          0.5f * xv * (1.f + erff(xv * 0.70710678118654752f));
    }
  }
  __syncthreads();
  // Restage W2 (64x512) into the W1 region, pad-516 rows (async)
  for (int t = tid; t < 64 * 128; t += 384) {
    const int dd = t >> 7, j = t & 127;
    g2lds16(w_c2 + dd * 512 + j * 4, sm + K3_WS + dd * 516 + j * 4);
  }
  lds_async_wait();
  __syncthreads();

  // GEMM2: Z[d][p] = sum_e W2[d][e]*T1[e][p]; out = BN2(Ht + Z + b_c2)^T
  {
    const int mt = wave & 3, nt = wave >> 2;
    const float* arow = sm + K3_WS + (mt * 16 + l16) * 516;
    v8f acc = {};
    #pragma unroll 8
    for (int kk = 0; kk < 512; kk += 4) {
      const int c0 = kk + (half ? 2 : 0);
      v2f a, bb;
      a.x = arow[c0]; a.y = arow[c0 + 1];
      bb.x = sm[K3_T1 + c0 * 48 + nt * 16 + l16];
      bb.y = sm[K3_T1 + (c0 + 1) * 48 + nt * 16 + l16];
      acc = wmma4(a, bb, acc);
    }
    const int pl = nt * 16 + l16;
    const int p  = p0 + pl;
    #pragma unroll
    for (int r = 0; r < 8; ++r) {
      const int dd = mt * 16 + r + 8 * half;
      const float z = acc[r] + b_c2[dd];
      const float rsum = sm[K3_HT + dd * 52 + pl] + z;
      const float s = g2[dd] * rsqrtf(v2[dd] + EPS_);
      out[((size_t)b * PRED_ + p) * 64 + dd] = (rsum - m2[dd]) * s + be2[dd];
    }
  }
}

extern "C" void kernel_launch(void* const* d_in, const int* in_sizes, int n_in,
                              void* d_out, int out_size, void* d_ws,
                              size_t ws_size, hipStream_t stream) {
  (void)in_sizes; (void)n_in; (void)out_size; (void)ws_size;
  const float* x      = (const float*)d_in[0];
  const float* x_date = (const float*)d_in[1];
  const float* y_date = (const float*)d_in[2];
  const float* w_lin  = (const float*)d_in[3];
  const float* b_lin  = (const float*)d_in[4];
  const float* w_c1   = (const float*)d_in[5];
  const float* b_c1   = (const float*)d_in[6];
  const float* w_c2   = (const float*)d_in[7];
  const float* b_c2   = (const float*)d_in[8];
  const float* g1     = (const float*)d_in[9];
  const float* be1    = (const float*)d_in[10];
  const float* m1     = (const float*)d_in[11];
  const float* v1     = (const float*)d_in[12];
  const float* g2     = (const float*)d_in[13];
  const float* be2    = (const float*)d_in[14];
  const float* m2     = (const float*)d_in[15];
  const float* v2     = (const float*)d_in[16];
  float* out   = (float*)d_out;
  float* vflat = (float*)d_ws;                       // 4096*1440 f32
  float* hbuf  = vflat + (size_t)4096 * OPL_;        // 4096*720 f32

  const size_t lds1 = (size_t)K1_TOT * sizeof(float);
  const size_t lds3 = (size_t)K3_TOT * sizeof(float);
  (void)hipFuncSetAttribute(reinterpret_cast<const void*>(attn_kernel),
                            hipFuncAttributeMaxDynamicSharedMemorySize, (int)lds1);
  (void)hipFuncSetAttribute(reinterpret_cast<const void*>(ffn_bn2_kernel),
                            hipFuncAttributeMaxDynamicSharedMemorySize, (int)lds3);

  attn_kernel<<<dim3(B_ * D_), 256, lds1, stream>>>(x, x_date, y_date, vflat);
  lin_bn1_kernel<<<dim3(64, 15), 384, 0, stream>>>(vflat, w_lin, b_lin, g1, be1,
                                                   m1, v1, hbuf);
  ffn_bn2_kernel<<<dim3(B_, 15), 384, lds3, stream>>>(hbuf, w_c1, b_c1, w_c2,
                                                      b_c2, g2, be2, m2, v2, out);
}